// Spline1DInterpolant_35476429865441
// MI455X (gfx1250) — compile-verified
//
#include <hip/hip_runtime.h>

// Cubic B-spline evaluation, 16.7M points, 4.2M-knot grid on [0,1).
// Bandwidth-bound gather kernel: c (16 MB) lives in L2 (192 MB); x/out are
// streamed with nontemporal hints so they don't evict it.

typedef float f32x4  __attribute__((ext_vector_type(4)));                 // 16B aligned
typedef float f32x4u __attribute__((ext_vector_type(4), aligned(4)));    // 4B aligned (gather)

#define NGRID 4194304
#define NC    (NGRID + 3)
#define SCALE 4194304.0f   // 1/h = n/(b-a)

__global__ __launch_bounds__(256) void
spline1d_eval_kernel(const float* __restrict__ x,
                     const float* __restrict__ c,
                     float* __restrict__ out,
                     int nvec, int npoints)
{
    const int tid    = blockIdx.x * blockDim.x + threadIdx.x;
    const int stride = gridDim.x * blockDim.x;

    const f32x4* __restrict__ xv4 = (const f32x4*)x;
    f32x4*       __restrict__ ov4 = (f32x4*)out;

    for (int v = tid; v < nvec; v += stride) {
        // Look ahead one grid-stride pass on the x stream (global_prefetch_b8).
        if (v + stride < nvec)
            __builtin_prefetch((const void*)(xv4 + v + stride), 0, 1);

        // Streaming load of 4 query points, nontemporal (TH=NT).
        f32x4 xq = __builtin_nontemporal_load(xv4 + v);

        f32x4 r;
#pragma unroll
        for (int j = 0; j < 4; ++j) {
            float s  = xq[j] * SCALE;          // (x - a)/h with a=0
            float bf = floorf(s);
            float f  = s - bf;                 // fractional position in cell
            int base = (int)bf;

            // Reference clips only the gather index (jnp.clip(i-1, 0, NC-1)).
            int gb = base;
            gb = gb < 0 ? 0 : gb;
            gb = gb > (NC - 4) ? (NC - 4) : gb;

            // One unaligned 16B gather: c[gb..gb+3] (L2-resident working set).
            f32x4u cv = *(const f32x4u*)(c + gb);

            // Four B-spline basis weights (taps i = base..base+3):
            //   |t| = 1+f, f, 1-f, 2-f  ->  (1-f)^3, inner(f), inner(1-f), f^3
            float g  = 1.0f - f;
            float w0 = g * g * g;                          // outer tap
            float w1 = 4.0f + f * f * (3.0f * f - 6.0f);   // inner tap
            float w2 = 4.0f + g * g * (3.0f * g - 6.0f);   // inner tap
            float w3 = f * f * f;                          // outer tap

            r[j] = w0 * cv[0] + w1 * cv[1] + w2 * cv[2] + w3 * cv[3];
        }

        // Streaming store, nontemporal.
        __builtin_nontemporal_store(r, ov4 + v);
    }

    // Scalar tail (npoints not divisible by 4). Never taken for 2^24 points,
    // but keeps the kernel correct for arbitrary sizes.
    int tail = npoints & 3;
    if (tail && tid < tail) {
        int idx  = nvec * 4 + tid;
        float s  = x[idx] * SCALE;
        float bf = floorf(s);
        float f  = s - bf;
        int gb   = (int)bf;
        gb = gb < 0 ? 0 : gb;
        gb = gb > (NC - 4) ? (NC - 4) : gb;
        float g  = 1.0f - f;
        float w0 = g * g * g;
        float w1 = 4.0f + f * f * (3.0f * f - 6.0f);
        float w2 = 4.0f + g * g * (3.0f * g - 6.0f);
        float w3 = f * f * f;
        out[idx] = w0 * c[gb] + w1 * c[gb + 1] + w2 * c[gb + 2] + w3 * c[gb + 3];
    }
}

extern "C" void kernel_launch(void* const* d_in, const int* in_sizes, int n_in,
                              void* d_out, int out_size, void* d_ws, size_t ws_size,
                              hipStream_t stream)
{
    const float* x = (const float*)d_in[0];   // [NPOINTS] query points
    const float* c = (const float*)d_in[1];   // [NC] spline coefficients
    float* out     = (float*)d_out;           // [NPOINTS]

    const int npoints = in_sizes[0];
    const int nvec    = npoints >> 2;

    // 256 threads = 8 wave32 waves per block; ~4 grid-stride passes per thread
    // so the x-stream prefetch has real lookahead. 4096 blocks saturates all
    // WGPs with deep VMEM queues.
    const int threads = 256;
    int blocks = (nvec + threads * 4 - 1) / (threads * 4);
    if (blocks < 1) blocks = 1;

    spline1d_eval_kernel<<<blocks, threads, 0, stream>>>(x, c, out, nvec, npoints);
}